// CapsuleNetwork_85555748537123
// MI455X (gfx1250) — compile-verified
//
#include <hip/hip_runtime.h>

// CapsuleNetwork forward for MI455X (gfx1250, wave32, WMMA).
// Phase 1: hat[b,s,k] = sum_d U[b,s,d] * W[s,k,d]  via V_WMMA_F32_16X16X4_F32,
//          2x2 register-blocked (32x32 per wave) for 2x operand reuse.
// Phase 2: 3-iteration dynamic routing per batch row, entirely in LDS (320KB/WGP).

typedef float v2f __attribute__((ext_vector_type(2)));
typedef float v8f __attribute__((ext_vector_type(8)));

#define B_    2048
#define S_    200
#define D_    64
#define NI_   4
#define KOUT  256   // NI*D
#define HSTR  65    // LDS row stride (floats) for hat: 64 + 1 pad -> conflict-free

// ---------------------------------------------------------------------------
// Phase 1: per-s GEMM  [B x D] @ [D x KOUT]  (Bmat(d,k) = W[s][k][d])
// One wave = one 32(M=batch) x 32(N=out-col) block = 2x2 WMMA tiles, K=64.
// 16 K-steps * 4 wmma = 64 wmma / wave; 4 b64 loads per lane per K-step.
// grid: 200 s * 64 Mblk * 8 Nblk = 102400 waves / 8 per block = 12800 blocks.
// ---------------------------------------------------------------------------
__global__ __launch_bounds__(256) void caps_gemm_wmma(
    const float* __restrict__ U,    // [B,S,D]
    const float* __restrict__ W,    // [S,KOUT,D]
    float* __restrict__ hat)        // [B,S,KOUT]  (workspace)
{
  const int lane = threadIdx.x & 31;
  const int wave = threadIdx.x >> 5;
  const int gw   = blockIdx.x * 8 + wave;          // 0 .. 102399
  const int nt   = gw & 7;                         // 8 N blocks of 32
  const int mt   = (gw >> 3) & 63;                 // 64 M blocks of 32
  const int s    = gw >> 9;                        // 0..199
  const int b0   = mt * 32;
  const int n0   = nt * 32;

  // ISA 7.12.2: 32-bit A 16x4: lanes 0-15 M=lane {K=0,1}, lanes 16-31 M=lane-16 {K=2,3}
  const int r     = lane & 15;                     // M for A, N for B
  const int khalf = (lane < 16) ? 0 : 2;

  const float* A0 = U + ((size_t)(b0 + r)      * S_ + s) * D_;   // + k
  const float* A1 = U + ((size_t)(b0 + 16 + r) * S_ + s) * D_;
  const float* B0 = W + ((size_t)s * KOUT + (n0 + r))      * D_; // + d
  const float* B1 = W + ((size_t)s * KOUT + (n0 + 16 + r)) * D_;

  v8f acc00 = {0.f,0.f,0.f,0.f,0.f,0.f,0.f,0.f};
  v8f acc01 = acc00, acc10 = acc00, acc11 = acc00;

#pragma unroll
  for (int kk = 0; kk < 16; ++kk) {
    const int k = kk * 4 + khalf;
    v2f a0; a0.x = A0[k]; a0.y = A0[k + 1];        // b64 loads per lane
    v2f a1; a1.x = A1[k]; a1.y = A1[k + 1];
    v2f w0; w0.x = B0[k]; w0.y = B0[k + 1];
    v2f w1; w1.x = B1[k]; w1.y = B1[k + 1];
    acc00 = __builtin_amdgcn_wmma_f32_16x16x4_f32(false, a0, false, w0, (short)0, acc00, false, false);
    acc01 = __builtin_amdgcn_wmma_f32_16x16x4_f32(false, a0, false, w1, (short)0, acc01, false, false);
    acc10 = __builtin_amdgcn_wmma_f32_16x16x4_f32(false, a1, false, w0, (short)0, acc10, false, false);
    acc11 = __builtin_amdgcn_wmma_f32_16x16x4_f32(false, a1, false, w1, (short)0, acc11, false, false);
  }

  // C layout: VGPR j -> row j (lanes 0-15) / row j+8 (lanes 16-31), col = lane&15.
  // hat is stream-once -> non-temporal stores keep W/U resident in L2.
  const int rowoff = (lane < 16) ? 0 : 8;
#pragma unroll
  for (int j = 0; j < 8; ++j) {
    const int row = j + rowoff;
    const size_t base0 = ((size_t)(b0 + row)      * S_ + s) * KOUT;
    const size_t base1 = ((size_t)(b0 + 16 + row) * S_ + s) * KOUT;
    __builtin_nontemporal_store(acc00[j], &hat[base0 + n0 + r]);
    __builtin_nontemporal_store(acc01[j], &hat[base0 + n0 + 16 + r]);
    __builtin_nontemporal_store(acc10[j], &hat[base1 + n0 + r]);
    __builtin_nontemporal_store(acc11[j], &hat[base1 + n0 + 16 + r]);
  }
}

// ---------------------------------------------------------------------------
// Phase 2: dynamic routing, one workgroup (256 thr = 8 waves) per batch row.
// Dynamic LDS layout (floats):
//   hat   [NI][S][HSTR]  52000
//   cw    [NI][S]          800   capsule_weight
//   sw    [NI][S]          800   masked softmax weights
//   inter [NI*D]           256   interest capsule
//   sq    [256]            256   reduction scratch
//   scl   [8]                8   per-capsule scale / logits / argmax
//   keep  [S]              200   mask
// total 54320 floats = 217280 B  (fits 320KB WGP LDS)
// ---------------------------------------------------------------------------
__global__ __launch_bounds__(256) void caps_route(
    const float* __restrict__ hat_g,   // [B,S,KOUT]
    const float* __restrict__ item_eb, // [B,D]
    const float* __restrict__ mask,    // [B,S]
    float* __restrict__ out)           // [B*NI*D] ++ [B*D]
{
  extern __shared__ float lds[];
  float* hat   = lds;                       // NI*S*HSTR
  float* cw    = hat + NI_ * S_ * HSTR;     // NI*S
  float* sw    = cw + NI_ * S_;             // NI*S
  float* inter = sw + NI_ * S_;             // 256
  float* sq    = inter + 256;               // 256
  float* scl   = sq + 256;                  // 8
  float* keep  = scl + 8;                   // S

  const int tid = threadIdx.x;
  const int b   = blockIdx.x;

  // Stage hat[b] into LDS with padded rows (read-once -> non-temporal).
  for (int idx = tid; idx < NI_ * S_ * D_; idx += 256) {
    const int n = idx / (S_ * D_);
    const int r = idx - n * (S_ * D_);
    const int s = r >> 6;
    const int d = r & 63;
    hat[(n * S_ + s) * HSTR + d] =
        __builtin_nontemporal_load(&hat_g[((size_t)b * S_ + s) * KOUT + n * D_ + d]);
  }
  for (int s = tid; s < S_; s += 256) keep[s] = mask[(size_t)b * S_ + s];
  for (int p = tid; p < NI_ * S_; p += 256) cw[p] = 0.f;
  __syncthreads();

  const int n_t = tid >> 6;   // capsule index 0..3
  const int d_t = tid & 63;   // dim index    0..63
  float capv = 0.f;

  for (int it = 0; it < 3; ++it) {
    // softmax over NI per position s, zeroed where mask==0
    for (int s = tid; s < S_; s += 256) {
      float m = cw[s];
#pragma unroll
      for (int n = 1; n < NI_; ++n) m = fmaxf(m, cw[n * S_ + s]);
      float e[NI_], sum = 0.f;
#pragma unroll
      for (int n = 0; n < NI_; ++n) { e[n] = __expf(cw[n * S_ + s] - m); sum += e[n]; }
      const float g = (keep[s] != 0.f) ? (1.f / sum) : 0.f;
#pragma unroll
      for (int n = 0; n < NI_; ++n) sw[n * S_ + s] = e[n] * g;
    }
    __syncthreads();

    // interest[n,d] = sum_s sw[n,s] * hat[n,s,d]   (thread t owns (n_t,d_t))
    float acc = 0.f;
    {
      const float* hrow = hat + n_t * S_ * HSTR + d_t;
      const float* swr  = sw + n_t * S_;
      for (int s = 0; s < S_; ++s) acc += swr[s] * hrow[s * HSTR];
    }

    // squash: scale[n] = (nn/(1+nn)) / sqrt(nn + 1e-9), nn = |interest[n]|^2
    sq[tid] = acc * acc;
    __syncthreads();
    if (tid < NI_) {
      float nn = 0.f;
      for (int d = 0; d < D_; ++d) nn += sq[tid * D_ + d];
      scl[tid] = (nn / (1.f + nn)) * rsqrtf(nn + 1e-9f);
    }
    __syncthreads();
    capv = scl[n_t] * acc;
    inter[tid] = capv;

    if (it < 2) {
      __syncthreads();
      // cw[n,s] += sum_d hat[n,s,d] * inter[n,d]
      for (int p = tid; p < NI_ * S_; p += 256) {
        const int n = p / S_;
        const int s = p - n * S_;
        const float* hr = hat + (n * S_ + s) * HSTR;
        const float* iv = inter + n * D_;
        float dsum = 0.f;
        for (int d = 0; d < D_; ++d) dsum += hr[d] * iv[d];
        cw[p] += dsum;
      }
      __syncthreads();
    }
  }
  __syncthreads();  // inter[] complete

  // output 0: interest_capsule [B,NI,D]
  out[(size_t)b * (NI_ * D_) + tid] = capv;

  // hard readout: argmax over softmax(logits) == argmax over logits
  sq[tid] = capv * item_eb[(size_t)b * D_ + d_t];
  __syncthreads();
  if (tid < NI_) {
    float lg = 0.f;
    for (int d = 0; d < D_; ++d) lg += sq[tid * D_ + d];
    scl[4 + tid] = lg;
  }
  __syncthreads();
  if (tid == 0) {
    int best = 0; float bv = scl[4];
#pragma unroll
    for (int n = 1; n < NI_; ++n)
      if (scl[4 + n] > bv) { bv = scl[4 + n]; best = n; }   // first-max tiebreak
    scl[0] = (float)best;
  }
  __syncthreads();
  if (tid < D_) {
    const int best = (int)scl[0];
    out[(size_t)B_ * NI_ * D_ + (size_t)b * D_ + tid] = inter[best * D_ + tid];
  }
}

// ---------------------------------------------------------------------------
extern "C" void kernel_launch(void* const* d_in, const int* in_sizes, int n_in,
                              void* d_out, int out_size, void* d_ws, size_t ws_size,
                              hipStream_t stream) {
  (void)in_sizes; (void)n_in; (void)out_size; (void)ws_size;
  const float* U    = (const float*)d_in[0];  // item_his_emb [B,S,D]
  const float* eb   = (const float*)d_in[1];  // item_eb      [B,D]
  const float* msk  = (const float*)d_in[2];  // mask         [B,S]
  const float* W    = (const float*)d_in[3];  // w            [1,S,NI*D,D]
  float* out = (float*)d_out;
  float* hat = (float*)d_ws;                  // needs B*S*KOUT*4 = 419,430,400 B

  // Phase 1: 200 s * 64 Mblk * 8 Nblk waves, 8 waves/block
  caps_gemm_wmma<<<dim3(S_ * 64 * 8 / 8), dim3(256), 0, stream>>>(U, W, hat);

  // Phase 2: one workgroup per batch row, 217,280 B dynamic LDS
  const size_t shmem = (size_t)(NI_ * S_ * HSTR + 2 * NI_ * S_ + 256 + 256 + 8 + S_)
                       * sizeof(float);
  caps_route<<<dim3(B_), dim3(256), shmem, stream>>>(hat, eb, msk, out);
}